// Proposal_Target_layer_91147795956373
// MI455X (gfx1250) — compile-verified
//
#include <hip/hip_runtime.h>
#include <hip/hip_bf16.h>
#include <math.h>

#ifndef __has_builtin
#define __has_builtin(x) 0
#endif

#if __has_builtin(__builtin_amdgcn_tensor_load_to_lds) && __has_builtin(__builtin_amdgcn_s_wait_tensorcnt)
#define USE_TDM 1
#else
#define USE_TDM 0
#endif

#define N_GT   128
#define N_FG   128
#define N_OUT  512
#define CHUNK  512
#define FG_TH  0.5f
#define BG_LO  0.1f
#define BG_HI  0.5f

typedef __attribute__((ext_vector_type(4))) unsigned int tdm_vu4;
typedef __attribute__((ext_vector_type(8))) int          tdm_vi8;
typedef __attribute__((ext_vector_type(4))) int          tdm_vi4;

// ---------------------------------------------------------------------------
// Kernel 1: per-ROI max IoU / argmax over GT boxes staged in LDS via TDM.
// Also emits fg/bg flags and per-chunk (512 ROIs) fg/bg counts.
// ---------------------------------------------------------------------------
__global__ void k_iou(const float4* __restrict__ rois, const float4* __restrict__ gt,
                      int n_prop, int n_all,
                      int* __restrict__ argmax, int* __restrict__ flags,
                      int* __restrict__ pfg, int* __restrict__ pbg) {
    __shared__ float4 gt_lds[N_GT];   // 2 KB GT tile, reused by 512 ROIs per block
    __shared__ float  gt_area[N_GT];
    __shared__ int    rfg[256], rbg[256];

    const int t = threadIdx.x;
    const int b = blockIdx.x;

#if USE_TDM
    // Stage the 2KB GT tile with the Tensor Data Mover (one DMA per block,
    // issued by wave 0; EXEC is ignored by TDM, the instruction issues once
    // per wave). Descriptor: 1-row tile of 512 f32 elements, contiguous.
    if (t < 32) {
        unsigned long long ga = (unsigned long long)(uintptr_t)gt;
        unsigned int lds_off  = (unsigned int)(uintptr_t)(&gt_lds[0]); // low 32b = LDS offset
        const unsigned nelem = N_GT * 4u;  // 512 f32 elements
        tdm_vu4 g0 = { 1u,                                    // count=1 (valid D#)
                       lds_off,                               // lds_addr
                       (unsigned)(ga & 0xFFFFFFFFull),        // global_addr[31:0]
                       (unsigned)((ga >> 32) & 0x1FFFFFFull)  // global_addr[56:32]
                           | (2u << 30) };                    // type=2 ("image")
        tdm_vi8 g1 = { (int)(2u << 16),          // data_size=4B code 2
                       (int)(nelem << 16),       // tensor_dim0[15:0] = 512
                       (int)(1u << 16),          // tensor_dim1 = 1
                       (int)(nelem << 16),       // tile_dim0 = 512
                       1,                        // tile_dim1 = 1
                       (int)nelem,               // tensor_dim0_stride = 512
                       (int)(nelem << 16),       // tensor_dim1_stride lo16
                       0 };
        tdm_vi4 gz = { 0, 0, 0, 0 };
        tdm_vi8 gx = { 0, 0, 0, 0, 0, 0, 0, 0 }; // unused upper descriptor group
        __builtin_amdgcn_tensor_load_to_lds(g0, g1, gz, gz, gx, 0);
        __builtin_amdgcn_s_wait_tensorcnt(0);   // s_wait_tensorcnt 0
    }
#else
    if (t < N_GT) gt_lds[t] = gt[t];
#endif
    __syncthreads();
    if (t < N_GT) {
        float4 g = gt_lds[t];
        gt_area[t] = (g.z - g.x + 1.0f) * (g.w - g.y + 1.0f);
    }
    __syncthreads();

    int cfg = 0, cbg = 0;
    #pragma unroll
    for (int k = 0; k < 2; ++k) {
        int i = b * CHUNK + t + k * 256;
        if (i < n_all) {
            float4 bx = (i < n_prop) ? rois[i] : gt[i - n_prop];
            float area_b = (bx.z - bx.x + 1.0f) * (bx.w - bx.y + 1.0f);
            float best = -1.0f;
            int   bgix = 0;
            #pragma unroll 8
            for (int g = 0; g < N_GT; ++g) {
                float4 gb = gt_lds[g];
                float iw = fminf(bx.z, gb.z) - fmaxf(bx.x, gb.x) + 1.0f;
                float ih = fminf(bx.w, gb.w) - fmaxf(bx.y, gb.y) + 1.0f;
                iw = fmaxf(iw, 0.0f);
                ih = fmaxf(ih, 0.0f);
                float inter = iw * ih;
                float v = inter / (area_b + gt_area[g] - inter);
                if (v > best) { best = v; bgix = g; }  // strict > keeps first occurrence
            }
            argmax[i] = bgix;
            int fgf = (best > FG_TH) ? 1 : 0;
            int bgf = (best > BG_LO && best < BG_HI) ? 1 : 0;
            flags[i] = fgf | (bgf << 1);
            cfg += fgf; cbg += bgf;
        }
    }
    rfg[t] = cfg; rbg[t] = cbg;
    __syncthreads();
    for (int s = 128; s > 0; s >>= 1) {
        if (t < s) { rfg[t] += rfg[t + s]; rbg[t] += rbg[t + s]; }
        __syncthreads();
    }
    if (t == 0) { pfg[b] = rfg[0]; pbg[b] = rbg[0]; }
}

// ---------------------------------------------------------------------------
// Kernel 2: single-block exclusive scan of chunk partials + sample-count meta.
// ---------------------------------------------------------------------------
__global__ void k_scan(const int* __restrict__ pfg, const int* __restrict__ pbg,
                       int* __restrict__ ofg, int* __restrict__ obg,
                       int* __restrict__ meta, int nblk) {
    __shared__ int sf[1024], sb[1024];
    const int t = threadIdx.x;
    int vf = (t < nblk) ? pfg[t] : 0;
    int vb = (t < nblk) ? pbg[t] : 0;
    sf[t] = vf; sb[t] = vb;
    __syncthreads();
    for (int off = 1; off < 1024; off <<= 1) {
        int af = 0, ab = 0;
        if (t >= off) { af = sf[t - off]; ab = sb[t - off]; }
        __syncthreads();
        sf[t] += af; sb[t] += ab;
        __syncthreads();
    }
    if (t < nblk) { ofg[t] = sf[t] - vf; obg[t] = sb[t] - vb; }
    if (t == 0) {
        int total_fg = sf[1023];
        int total_bg = sb[1023];
        int n_fg   = (total_fg < N_FG) ? total_fg : N_FG;
        int cap_bg = N_OUT - n_fg;
        int n_bg   = (total_bg < cap_bg) ? total_bg : cap_bg;
        meta[0] = n_fg; meta[1] = cap_bg; meta[2] = n_bg;
    }
}

// ---------------------------------------------------------------------------
// Kernel 3: reconstruct global cumsums; stable 3-way-partition positions.
// key = 0 (selected fg), 1 (selected bg), 2 (rest); stable argsort prefix
// == closed-form scatter position per ROI. Write idx for pos < 512.
// ---------------------------------------------------------------------------
__device__ __forceinline__ void emit_idx(int i, int fgf, int bgf, int fg_cum, int bg_cum,
                                         int n_fg, int cap_bg, int n_bg,
                                         int* __restrict__ idx_out) {
    int pos;
    if (fgf && fg_cum <= N_FG) {
        pos = fg_cum - 1;
    } else if (bgf && bg_cum <= cap_bg) {
        pos = n_fg + bg_cum - 1;
    } else {
        int selfg = (fg_cum < N_FG) ? fg_cum : N_FG;
        int selbg = (bg_cum < cap_bg) ? bg_cum : cap_bg;
        int other_cum = (i + 1) - selfg - selbg;
        pos = n_fg + n_bg + other_cum - 1;
    }
    if (pos >= 0 && pos < N_OUT) idx_out[pos] = i;
}

__global__ void k_select(const int* __restrict__ flags,
                         const int* __restrict__ ofg, const int* __restrict__ obg,
                         const int* __restrict__ meta, int n_all,
                         int* __restrict__ idx_out) {
    __shared__ int sf[256], sb[256];
    const int t = threadIdx.x;
    const int b = blockIdx.x;
    const int i0 = b * CHUNK + 2 * t;
    const int i1 = i0 + 1;
    int f0 = (i0 < n_all) ? flags[i0] : 0;
    int f1 = (i1 < n_all) ? flags[i1] : 0;
    int fg0 = f0 & 1, bg0 = (f0 >> 1) & 1;
    int fg1 = f1 & 1, bg1 = (f1 >> 1) & 1;
    sf[t] = fg0 + fg1; sb[t] = bg0 + bg1;
    __syncthreads();
    for (int off = 1; off < 256; off <<= 1) {
        int af = 0, ab = 0;
        if (t >= off) { af = sf[t - off]; ab = sb[t - off]; }
        __syncthreads();
        sf[t] += af; sb[t] += ab;
        __syncthreads();
    }
    int exf = sf[t] - (fg0 + fg1);   // exclusive prefix of this thread's pair
    int exb = sb[t] - (bg0 + bg1);
    int gfg = ofg[b] + exf;          // fg count strictly before i0
    int gbg = obg[b] + exb;
    int n_fg = meta[0], cap_bg = meta[1], n_bg = meta[2];
    if (i0 < n_all)
        emit_idx(i0, fg0, bg0, gfg + fg0, gbg + bg0, n_fg, cap_bg, n_bg, idx_out);
    if (i1 < n_all)
        emit_idx(i1, fg1, bg1, gfg + fg0 + fg1, gbg + bg0 + bg1, n_fg, cap_bg, n_bg, idx_out);
}

// ---------------------------------------------------------------------------
// Kernel 4: gather 512 selected ROIs, bbox transform, normalize, labels.
// d_out layout: rois_out[512*4] | delta[512*4] | labels[512] (int32 bits).
// ---------------------------------------------------------------------------
__global__ void k_output(const float4* __restrict__ rois, const float4* __restrict__ gt,
                         const int* __restrict__ labels, const int* __restrict__ argmax,
                         const int* __restrict__ idx_out, int n_prop,
                         float* __restrict__ out) {
    int j = blockIdx.x * blockDim.x + threadIdx.x;
    if (j >= N_OUT) return;
    int i = idx_out[j];
    float4 r = (i < n_prop) ? rois[i] : gt[i - n_prop];
    int g = argmax[i];
    float4 gb = gt[g];
    float w  = r.z - r.x + 1.0f, h  = r.w - r.y + 1.0f;
    float cx = r.x + 0.5f * w,   cy = r.y + 0.5f * h;
    float gw  = gb.z - gb.x + 1.0f, gh  = gb.w - gb.y + 1.0f;
    float gcx = gb.x + 0.5f * gw,   gcy = gb.y + 0.5f * gh;
    const float inv_std = 5.0f;  // (d - 0) / 0.2
    float4 d;
    d.x = ((gcx - cx) / w) * inv_std;
    d.y = ((gcy - cy) / h) * inv_std;
    d.z = logf(gw / w) * inv_std;
    d.w = logf(gh / h) * inv_std;
    ((float4*)out)[j] = r;
    ((float4*)(out + 4 * N_OUT))[j] = d;
    ((int*)(out + 8 * N_OUT))[j] = labels[g];
}

// ---------------------------------------------------------------------------
extern "C" void kernel_launch(void* const* d_in, const int* in_sizes, int n_in,
                              void* d_out, int out_size, void* d_ws, size_t ws_size,
                              hipStream_t stream) {
    (void)n_in; (void)out_size; (void)ws_size;
    const float4* rois   = (const float4*)d_in[0];
    const float4* gt     = (const float4*)d_in[1];
    const int*    labels = (const int*)d_in[2];

    const int n_prop = in_sizes[0] / 4;          // 262144
    const int n_gt   = in_sizes[1] / 4;          // 128
    const int n_all  = n_prop + n_gt;            // 262272
    const int nblk   = (n_all + CHUNK - 1) / CHUNK;  // 513

    char* w = (char*)d_ws;
    int* argmax  = (int*)w; w += (size_t)n_all * sizeof(int);
    int* flags   = (int*)w; w += (size_t)n_all * sizeof(int);
    int* pfg     = (int*)w; w += (size_t)nblk * sizeof(int);
    int* pbg     = (int*)w; w += (size_t)nblk * sizeof(int);
    int* ofg     = (int*)w; w += (size_t)nblk * sizeof(int);
    int* obg     = (int*)w; w += (size_t)nblk * sizeof(int);
    int* meta    = (int*)w; w += 64;
    int* idx_out = (int*)w; w += N_OUT * sizeof(int);

    k_iou<<<nblk, 256, 0, stream>>>(rois, gt, n_prop, n_all, argmax, flags, pfg, pbg);
    k_scan<<<1, 1024, 0, stream>>>(pfg, pbg, ofg, obg, meta, nblk);
    k_select<<<nblk, 256, 0, stream>>>(flags, ofg, obg, meta, n_all, idx_out);
    k_output<<<(N_OUT + 255) / 256, 256, 0, stream>>>(rois, gt, labels, argmax, idx_out,
                                                      n_prop, (float*)d_out);
}